// FeatureEmbeddingBank_163208757437
// MI455X (gfx1250) — compile-verified
//
#include <hip/hip_runtime.h>

// ---- problem constants (from reference) ----
#define N_SINGLE      24
#define N_MULTI       8
#define MULTI_LEN     20
#define N_FEAT        32          // 24 + 8
#define VOCAB         100000
#define ROWS_PER_FEAT 100001      // VOCAB + 1
#define EMB           64
#define TOTAL_W       184         // 24 + 8*20

// ---- task decomposition: one wave32 per task, 11 tasks per batch row ----
#define SINGLES_PER_TASK 8
#define SINGLE_TASKS     3        // 3 * 8 = 24 single features
#define TASKS_PER_BATCH  (SINGLE_TASKS + N_MULTI)  // 11
#define WAVES_PER_BLOCK  8        // 256 threads

typedef __attribute__((ext_vector_type(2))) float v2f;

__global__ __launch_bounds__(256) void
FeatureEmbeddingBank_kernel(const int* __restrict__ feats,
                            const float* __restrict__ W,
                            float* __restrict__ out,
                            int batch)
{
    const int lane = threadIdx.x & 31;
    // Force wave-uniformity so index fetches lower to scalar (s_load) reads.
    const int wave_in_blk = __builtin_amdgcn_readfirstlane((int)(threadIdx.x >> 5));
    const int task_g = blockIdx.x * WAVES_PER_BLOCK + wave_in_blk;

    const int b = task_g / TASKS_PER_BATCH;
    const int t = task_g - b * TASKS_PER_BATCH;
    if (b >= batch) return;

    const int*   fb = feats + (size_t)b * TOTAL_W;          // uniform pointer
    float*       ob = out   + (size_t)b * (N_FEAT * EMB);

    if (t < SINGLE_TASKS) {
        // ---- 8 single-token features: pure row gather, 1 load + 1 store each
        const int f0 = t * SINGLES_PER_TASK;
#pragma unroll
        for (int i = 0; i < SINGLES_PER_TASK; ++i) {
            const int f = f0 + i;
            int idx = fb[f];                                // s_load (uniform)
            idx = idx < 0 ? 0 : (idx > VOCAB ? VOCAB : idx);
            const v2f* row =
                (const v2f*)(W + ((size_t)f * ROWS_PER_FEAT + (size_t)idx) * EMB);
            v2f v = row[lane];                              // global_load_b64, coalesced
            // NT store: output is write-once; keep it out of L2 so W stays resident.
            __builtin_nontemporal_store(v, ((v2f*)(ob + f * EMB)) + lane);
        }
    } else {
        // ---- one multi-token feature: mean of 20 gathered rows
        const int m = t - SINGLE_TASKS;
        const int f = N_SINGLE + m;
        const int* mi = fb + N_SINGLE + m * MULTI_LEN;      // 20 uniform indices
        const float* Wf = W + (size_t)f * ROWS_PER_FEAT * EMB;

        v2f acc; acc.x = 0.0f; acc.y = 0.0f;
        int cnt = 0;
#pragma unroll
        for (int j = 0; j < MULTI_LEN; ++j) {
            int idx = mi[j];                                // s_load (uniform, batched)
            idx = idx < 0 ? 0 : (idx > VOCAB ? VOCAB : idx);
            cnt += (idx != 0) ? 1 : 0;                      // scalar per wave
            // 20 independent global_load_b64 — compiler issues all before waiting,
            // giving ~20 outstanding loads per wave to saturate HBM.
            v2f v = ((const v2f*)(Wf + (size_t)idx * EMB))[lane];
            acc.x += v.x;
            acc.y += v.y;
        }
        const float cntf = (float)(cnt > 0 ? cnt : 1);
        acc.x = acc.x / cntf;                               // exact f32 div (match ref)
        acc.y = acc.y / cntf;
        __builtin_nontemporal_store(acc, ((v2f*)(ob + f * EMB)) + lane);
    }
}

extern "C" void kernel_launch(void* const* d_in, const int* in_sizes, int n_in,
                              void* d_out, int out_size, void* d_ws, size_t ws_size,
                              hipStream_t stream)
{
    const int*   feats = (const int*)d_in[0];    // (B, 184) int32
    const float* W     = (const float*)d_in[1];  // (32, 100001, 64) f32
    float*       out   = (float*)d_out;          // (B, 32, 64) f32

    const int batch = in_sizes[0] / TOTAL_W;     // 4096
    const int total_tasks = batch * TASKS_PER_BATCH;
    const int blocks = (total_tasks + WAVES_PER_BLOCK - 1) / WAVES_PER_BLOCK;

    FeatureEmbeddingBank_kernel<<<blocks, 256, 0, stream>>>(feats, W, out, batch);
}